// EASTPostProcessor_38792144617504
// MI455X (gfx1250) — compile-verified
//
#include <hip/hip_runtime.h>
#include <cstdint>

// EAST post-processor for MI455X (gfx1250, wave32).
//  Kernel 1 : exact masked top-1000/image via LDS radix-select + bitonic sort,
//             then gather+decode only the winning boxes (never materializes
//             the 75 MB decoded-box tensor).
//  Kernel 2a: suppression bitmask build, spread over 256 blocks (machine-wide);
//             boxes staged to LDS with CDNA5 async DMA.
//  Kernel 2b: serial greedy NMS scan, one wave32/image, keep-bits in registers;
//             mask chunks async-DMA'd global->LDS (ASYNCcnt).

#define HW_     (384 * 384)
#define W_      384
#define KTOP    1000
#define NIMG    32
#define NTHR    1024
#define THRESH  0.05f
#define NEGV    -1000000000.0f
#define NMS_T   0.4f
#define CH      250   // scan-chunk rows (4 * 250 = 1000), 32 KB of mask in LDS
#define RPB     125   // mask-build rows per block (8 * 125 = 1000)

// monotonic float -> uint key (descending float == descending key)
__device__ __forceinline__ unsigned keyOf(float s) {
  float m = (s > THRESH) ? s : NEGV;
  unsigned u = __float_as_uint(m);
  return (u & 0x80000000u) ? ~u : (u | 0x80000000u);
}
__device__ __forceinline__ float keyToFloat(unsigned k) {
  unsigned u = (k & 0x80000000u) ? (k ^ 0x80000000u) : ~k;
  return __uint_as_float(u);
}

__device__ __forceinline__ void asyncLoadB128(unsigned ldsAddr, const void* gptr) {
  unsigned long long ga = (unsigned long long)(uintptr_t)gptr;
  asm volatile("global_load_async_to_lds_b128 %0, %1, off"
               :: "v"(ldsAddr), "v"(ga) : "memory");
}
__device__ __forceinline__ void waitAsync0() {
  asm volatile("s_wait_asynccnt 0x0" ::: "memory");
}
__device__ __forceinline__ unsigned ldsAddrOf(const void* p) {
  return (unsigned)(uintptr_t)p;  // low 32 bits of flat shared addr = LDS addr
}

// ---------------------------------------------------------------------------
// Kernel 1: exact top-1000 (sorted desc, lowest-index ties) + decode winners
// ---------------------------------------------------------------------------
__global__ __launch_bounds__(NTHR) void topk_decode_kernel(
    const float* __restrict__ cls, const float* __restrict__ reg,
    float* __restrict__ boxesK, float* __restrict__ scoresK) {
  const int n = blockIdx.x;
  const int tid = threadIdx.x;
  const float* c = cls + (size_t)n * HW_;

  __shared__ unsigned hist[256];
  __shared__ unsigned sPrefix, sNeeded, sCntEq, sIT, sTakeAll, sCnt;
  __shared__ unsigned long long cand[NTHR];

  if (tid == 0) { sNeeded = KTOP; sPrefix = 0; }

  // ---- 4-pass 8-bit radix select: exact 32-bit key threshold T ----
  for (int p = 0; p < 4; ++p) {
    for (int b = tid; b < 256; b += NTHR) hist[b] = 0;
    __syncthreads();
    const unsigned pref = sPrefix;
    const unsigned need = sNeeded;
    const int shift = 24 - 8 * p;
    for (int i = tid * 4; i < HW_; i += NTHR * 4) {
      float4 v = *reinterpret_cast<const float4*>(c + i);
      float ss[4] = {v.x, v.y, v.z, v.w};
#pragma unroll
      for (int q = 0; q < 4; ++q) {
        unsigned k = keyOf(ss[q]);
        unsigned hi = (p == 0) ? 0u : (k >> (32 - 8 * p));
        if (hi == pref) atomicAdd(&hist[(k >> shift) & 0xFF], 1u);
      }
    }
    __syncthreads();
    if (tid == 0) {
      unsigned cum = 0;
      for (int b = 255; b >= 0; --b) {
        unsigned h = hist[b];
        if (cum + h >= need) {
          sNeeded = need - cum; sCntEq = h;
          sPrefix = (pref << 8) | (unsigned)b;
          break;
        }
        cum += h;
      }
    }
    __syncthreads();
  }
  const unsigned T = sPrefix;
  const unsigned needEq = sNeeded;
  const unsigned cntEq = sCntEq;

  if (tid == 0) { sTakeAll = (cntEq == needEq) ? 1u : 0u; sIT = 0;
                  sNeeded = needEq; sPrefix = 0; }
  __syncthreads();

  if (!sTakeAll) {
    // tie-break among key==T by smallest index (matches lax.top_k)
    for (int p = 0; p < 3; ++p) {
      for (int b = tid; b < 64; b += NTHR) hist[b] = 0;
      __syncthreads();
      const unsigned pref = sPrefix;
      const unsigned need = sNeeded;
      const int shift = 12 - 6 * p;
      for (int i = tid * 4; i < HW_; i += NTHR * 4) {
        float4 v = *reinterpret_cast<const float4*>(c + i);
        float ss[4] = {v.x, v.y, v.z, v.w};
#pragma unroll
        for (int q = 0; q < 4; ++q) {
          if (keyOf(ss[q]) != T) continue;
          unsigned idx = (unsigned)(i + q);
          if ((idx >> (18 - 6 * p)) == pref)
            atomicAdd(&hist[(idx >> shift) & 0x3F], 1u);
        }
      }
      __syncthreads();
      if (tid == 0) {
        unsigned cum = 0;
        for (int b = 0; b < 64; ++b) {
          unsigned h = hist[b];
          if (cum + h >= need) {
            sNeeded = need - cum;
            sPrefix = (pref << 6) | (unsigned)b;
            break;
          }
          cum += h;
        }
      }
      __syncthreads();
    }
    if (tid == 0) sIT = sPrefix;
    __syncthreads();
  }
  const bool takeAll = (sTakeAll != 0);
  const unsigned IT = sIT;

  // ---- compact exactly KTOP (key, ~idx) pairs into LDS ----
  if (tid == 0) sCnt = 0;
  __syncthreads();
  for (int i = tid * 4; i < HW_; i += NTHR * 4) {
    float4 v = *reinterpret_cast<const float4*>(c + i);
    float ss[4] = {v.x, v.y, v.z, v.w};
#pragma unroll
    for (int q = 0; q < 4; ++q) {
      unsigned k = keyOf(ss[q]);
      unsigned idx = (unsigned)(i + q);
      bool sel = (k > T) || ((k == T) && (takeAll || idx <= IT));
      if (sel) {
        unsigned pos = atomicAdd(&sCnt, 1u);
        if (pos < NTHR)
          cand[pos] = ((unsigned long long)k << 32) | (unsigned)(~idx);
      }
    }
  }
  __syncthreads();
  if (tid >= (int)sCnt) cand[tid] = 0ull;  // pad to 1024
  __syncthreads();

  // ---- bitonic sort (descending): key desc, index asc on ties ----
  for (unsigned kk = 2; kk <= NTHR; kk <<= 1) {
    for (unsigned j = kk >> 1; j > 0; j >>= 1) {
      unsigned i = (unsigned)tid, ixj = i ^ j;
      if (ixj > i) {
        unsigned long long a = cand[i], b = cand[ixj];
        bool descBlk = ((i & kk) == 0);
        if (descBlk ? (a < b) : (a > b)) { cand[i] = b; cand[ixj] = a; }
      }
      __syncthreads();
    }
  }

  // ---- gather + decode only the KTOP winners ----
  if (tid < KTOP) {
    unsigned long long e = cand[tid];
    unsigned k = (unsigned)(e >> 32);
    unsigned idx = ~(unsigned)e;
    float score = keyToFloat(k);
    unsigned px = idx % W_, py = idx / W_;
    const float* r = reg + (size_t)n * 4 * HW_;
    float d0 = r[idx];
    float d1 = r[HW_ + idx];
    float d2 = r[2 * HW_ + idx];
    float d3 = r[3 * HW_ + idx];
    float x = (float)px * 4.0f, y = (float)py * 4.0f;
    float4 bx = make_float4(x - d3, y - d0, x + d1, y + d2);
    *reinterpret_cast<float4*>(boxesK + ((size_t)n * KTOP + tid) * 4) = bx;
    scoresK[(size_t)n * KTOP + tid] = score;
  }
}

// ---------------------------------------------------------------------------
// Kernel 2a: suppression bitmask build (machine-wide parallel)
// grid = (NIMG, KTOP/RPB); block (n, g) computes rows [g*RPB, (g+1)*RPB)
// gmask layout: [n][i][wc], bit b of word wc set iff iou(i, wc*32+b)>thr, j>i
// ---------------------------------------------------------------------------
__global__ __launch_bounds__(NTHR) void nms_mask_kernel(
    const float* __restrict__ boxesK, unsigned* __restrict__ gmask) {
  const int n = blockIdx.x;
  const int g = blockIdx.y;
  const int tid = threadIdx.x;
  __shared__ __align__(16) float sbox[KTOP * 4];
  __shared__ float sarea[KTOP];

  if (tid < KTOP)
    asyncLoadB128(ldsAddrOf(&sbox[tid * 4]),
                  boxesK + ((size_t)n * KTOP + tid) * 4);
  waitAsync0();
  __syncthreads();
  if (tid < KTOP) {
    float w = sbox[tid * 4 + 2] - sbox[tid * 4 + 0];
    float h = sbox[tid * 4 + 3] - sbox[tid * 4 + 1];
    sarea[tid] = fmaxf(w, 0.0f) * fmaxf(h, 0.0f);
  }
  __syncthreads();

  const int r0 = g * RPB;
  for (int wd = tid; wd < RPB * 32; wd += NTHR) {
    int wc = wd / RPB;          // word column (same for a whole wave mostly)
    int r = wd % RPB;
    int i = r0 + r;
    unsigned m = 0;
    if (wc * 32 + 31 > i) {     // skip words entirely at/below the diagonal
      float x1 = sbox[i * 4 + 0], y1 = sbox[i * 4 + 1];
      float x2 = sbox[i * 4 + 2], y2 = sbox[i * 4 + 3];
      float ai = sarea[i];
      int jbase = wc * 32;
#pragma unroll 4
      for (int b = 0; b < 32; ++b) {
        int j = jbase + b;
        if (j > i && j < KTOP) {
          float ix1 = fmaxf(x1, sbox[j * 4 + 0]);
          float iy1 = fmaxf(y1, sbox[j * 4 + 1]);
          float ix2 = fminf(x2, sbox[j * 4 + 2]);
          float iy2 = fminf(y2, sbox[j * 4 + 3]);
          float inter = fmaxf(ix2 - ix1, 0.0f) * fmaxf(iy2 - iy1, 0.0f);
          float uni = ai + sarea[j] - inter;
          float iou = inter / fmaxf(uni, 1e-9f);
          if (iou > NMS_T) m |= (1u << b);
        }
      }
    }
    gmask[(size_t)n * (KTOP * 32) + (size_t)i * 32 + wc] = m;
  }
}

// ---------------------------------------------------------------------------
// Kernel 2b: serial greedy scan (one wave32 per image) + output emit
// ---------------------------------------------------------------------------
__global__ __launch_bounds__(NTHR) void nms_scan_kernel(
    const float* __restrict__ boxesK, const float* __restrict__ scoresK,
    const unsigned* __restrict__ gmask, float* __restrict__ out) {
  const int n = blockIdx.x;
  const int tid = threadIdx.x;
  __shared__ __align__(16) float sbox[KTOP * 4];
  __shared__ float sscore[KTOP];
  __shared__ __align__(16) unsigned smaskc[CH * 32];  // 32 KB chunk
  __shared__ unsigned skeep[32];

  if (tid < KTOP) {
    asyncLoadB128(ldsAddrOf(&sbox[tid * 4]),
                  boxesK + ((size_t)n * KTOP + tid) * 4);
    sscore[tid] = scoresK[(size_t)n * KTOP + tid];
  }
  waitAsync0();
  __syncthreads();

  // validity; pack keep-init words via wave32 ballot
  bool valid = false;
  if (tid < KTOP) {
    float w = sbox[tid * 4 + 2] - sbox[tid * 4 + 0];
    float h = sbox[tid * 4 + 3] - sbox[tid * 4 + 1];
    valid = (sscore[tid] > THRESH) && (w >= 0.0f) && (h >= 0.0f);
  }
  unsigned long long bal = __ballot(valid);
  if ((tid & 31) == 0) skeep[tid >> 5] = (unsigned)bal;
  __syncthreads();

  unsigned keepw = (tid < 32) ? skeep[tid] : 0u;

  const unsigned* gmn = gmask + (size_t)n * (KTOP * 32);
  for (int c0 = 0; c0 < KTOP; c0 += CH) {
    __syncthreads();  // previous chunk's scan complete before overwrite
    // async-copy CH rows of mask (32 KB) global -> LDS
    for (int t = tid; t < CH * 32 / 4; t += NTHR)
      asyncLoadB128(ldsAddrOf(&smaskc[t * 4]), gmn + c0 * 32 + t * 4);
    waitAsync0();
    __syncthreads();
    // greedy sequential scan: lane w owns keep word w in a register
    if (tid < 32) {
      for (int r = 0; r < CH; ++r) {
        int i = c0 + r;
        int wsrc = i >> 5, bit = i & 31;
        unsigned kw = __shfl(keepw, wsrc, 32);
        if ((kw >> bit) & 1u) keepw &= ~smaskc[r * 32 + tid];
      }
    }
  }
  __syncthreads();
  if (tid < 32) skeep[tid] = keepw;
  __syncthreads();

  // emit out=(N,K,5)*keep then keep=(N,K) as floats
  if (tid < KTOP) {
    unsigned keep = (skeep[tid >> 5] >> (tid & 31)) & 1u;
    float f = keep ? 1.0f : 0.0f;
    size_t ob = (size_t)n * KTOP + tid;
    out[ob * 5 + 0] = sbox[tid * 4 + 0] * f;
    out[ob * 5 + 1] = sbox[tid * 4 + 1] * f;
    out[ob * 5 + 2] = sbox[tid * 4 + 2] * f;
    out[ob * 5 + 3] = sbox[tid * 4 + 3] * f;
    out[ob * 5 + 4] = sscore[tid] * f;
    out[(size_t)NIMG * KTOP * 5 + ob] = f;
  }
}

extern "C" void kernel_launch(void* const* d_in, const int* in_sizes, int n_in,
                              void* d_out, int out_size, void* d_ws, size_t ws_size,
                              hipStream_t stream) {
  const float* box_cls = (const float*)d_in[0];         // (32,1,384,384) f32
  const float* box_reg = (const float*)d_in[1];         // (32,4,384,384) f32
  float* out = (float*)d_out;                           // 160000 + 32000 floats
  float* boxesK = (float*)d_ws;                         // N*K*4 floats (512 KB)
  float* scoresK = boxesK + (size_t)NIMG * KTOP * 4;    // N*K floats  (128 KB)
  unsigned* gmask = (unsigned*)(scoresK + (size_t)NIMG * KTOP);  // N*K*32 words (4 MB)

  topk_decode_kernel<<<NIMG, NTHR, 0, stream>>>(box_cls, box_reg, boxesK, scoresK);
  nms_mask_kernel<<<dim3(NIMG, KTOP / RPB), NTHR, 0, stream>>>(boxesK, gmask);
  nms_scan_kernel<<<NIMG, NTHR, 0, stream>>>(boxesK, scoresK, gmask, out);

  (void)in_sizes; (void)n_in; (void)out_size; (void)ws_size;
}